// NFG_Prototypical_Nets_55834574848399
// MI455X (gfx1250) — compile-verified
//
#include <hip/hip_runtime.h>
#include <hip/hip_bf16.h>
#include <math.h>

typedef __attribute__((ext_vector_type(2))) float v2f;
typedef __attribute__((ext_vector_type(8))) float v8f;

#define N_WAY    64
#define N_SHOT   16
#define N_QUERY  512
#define DIM      2048
#define NQ_TOTAL (N_WAY * N_QUERY)       // 32768
#define QT       16                      // queries per block tile
#define KC       512                     // K chunk staged in LDS
#define LDS_STRIDE 516                   // 512 + 4 pad -> conflict-free A reads
#define OUT_SCALARS (N_WAY * N_QUERY * N_WAY)  // 2097152
#define INV_D (1.0f / 2048.0f)
#define INV_N (1.0f / 32768.0f)

// ---------------------------------------------------------------------------
// Swizzled B-fragment index for V_WMMA_F32_16X16X4_F32:
//   B is 4x16 (KxN). Assumed layout (mirrors documented A 16x4 layout):
//   lanes 0-15 hold K=0,1 (VGPR0,VGPR1), lanes 16-31 hold K=2,3; N = lane%16.
//   Stored so each lane's two floats are contiguous -> global_load_b64.
// ---------------------------------------------------------------------------
__device__ __forceinline__ int protoB_index(int n, int k) {
    const int t    = n >> 4;        // 16-proto tile (0..3)
    const int nl   = n & 15;
    const int step = k >> 2;        // k-step (0..511)
    const int kk   = k & 3;
    const int lane = (kk < 2) ? nl : (16 + nl);
    const int slot = kk & 1;
    return ((t * 512 + step) * 32 + lane) * 2 + slot;
}

// ---------------------------------------------------------------------------
// Zero the atomic accumulators (d_out scalar slots + |proto|^2 workspace).
// ---------------------------------------------------------------------------
__global__ void init_kernel(float* __restrict__ out, float* __restrict__ sqp) {
    const int t = threadIdx.x;
    if (t < N_WAY) sqp[t] = 0.0f;
    if (t == 0) {
        out[OUT_SCALARS]     = 0.0f;   // ce_loss
        out[OUT_SCALARS + 1] = 0.0f;   // acc
    }
}

// ---------------------------------------------------------------------------
// Prototypes: mean over 16 shots -> swizzled WMMA-B layout + |p|^2 per way.
// 512 blocks x 256 threads; 8 blocks per way (all threads in a block share n).
// ---------------------------------------------------------------------------
__global__ void __launch_bounds__(256)
proto_kernel(const float* __restrict__ s_emb,
             float* __restrict__ protoB, float* __restrict__ sqp) {
    const int n = blockIdx.x >> 3;
    const int k = ((blockIdx.x & 7) << 8) + threadIdx.x;

    const float* p = s_emb + (size_t)n * (N_SHOT * DIM) + k;
    float s = 0.0f;
#pragma unroll
    for (int sh = 0; sh < N_SHOT; ++sh) s += p[(size_t)sh * DIM];
    const float m = s * (1.0f / (float)N_SHOT);

    protoB[protoB_index(n, k)] = m;

    // block-reduce m*m -> one atomic per block
    float v = m * m;
    for (int o = 16; o > 0; o >>= 1) v += __shfl_xor(v, o, 32);
    __shared__ float wsum[8];
    const int lane = threadIdx.x & 31, w = threadIdx.x >> 5;
    if (lane == 0) wsum[w] = v;
    __syncthreads();
    if (threadIdx.x == 0) {
        float t = 0.0f;
#pragma unroll
        for (int i = 0; i < 8; ++i) t += wsum[i];
        atomicAdd(&sqp[n], t);
    }
}

// ---------------------------------------------------------------------------
// Main fused kernel: 16-query tile x 64 protos GEMM (f32 WMMA) + log-softmax
// + CE/acc reduction. 2048 blocks x 128 threads (4 waves, 1 proto-tile each).
// ---------------------------------------------------------------------------
__global__ void __launch_bounds__(128)
proto_dist_kernel(const float* __restrict__ q_emb,
                  const float* __restrict__ protoB,
                  const float* __restrict__ sqp,
                  float* __restrict__ out) {
    __shared__ __align__(16) float qtile[QT * LDS_STRIDE];  // 33,024 B
    __shared__ float distT[QT * N_WAY];                     //  4,096 B
    __shared__ float sqq[QT];
    __shared__ float blockCE, blockAC;

    const int tid  = threadIdx.x;
    const int lane = tid & 31;
    const int wave = tid >> 5;                 // proto tile 0..3
    const int qt   = blockIdx.x;               // query tile
    const size_t qbase = (size_t)qt * QT * DIM;

    if (tid < QT) sqq[tid] = 0.0f;
    if (tid == 0) { blockCE = 0.0f; blockAC = 0.0f; }

    v8f acc = {};
    float sqacc[QT];
#pragma unroll
    for (int i = 0; i < QT; ++i) sqacc[i] = 0.0f;

    // A-fragment addressing: lane l reads row (l%16), K-pair offset 0 or 2
    const int arow  = lane & 15;
    const int akofs = (lane >> 4) << 1;

    for (int chunk = 0; chunk < DIM / KC; ++chunk) {
        __syncthreads();
        // Stage 16 x 512 f32 (coalesced float4) + accumulate |q|^2 partials
#pragma unroll
        for (int i = 0; i < QT; ++i) {
            const float4 v = *(const float4*)(q_emb + qbase + (size_t)i * DIM
                                              + (size_t)chunk * KC + tid * 4);
            *(float4*)(&qtile[i * LDS_STRIDE + tid * 4]) = v;
            sqacc[i] += v.x * v.x + v.y * v.y + v.z * v.z + v.w * v.w;
        }
        __syncthreads();

        // 128 K-steps of V_WMMA_F32_16X16X4_F32
        const float* bp = protoB
            + ((size_t)(wave * 512 + chunk * (KC / 4)) * 32 + lane) * 2;
#pragma unroll 4
        for (int step = 0; step < KC / 4; ++step) {
            v2f a = *(const v2f*)(&qtile[arow * LDS_STRIDE + step * 4 + akofs]);
            v2f b = *(const v2f*)(bp + (size_t)step * 64);
            acc = __builtin_amdgcn_wmma_f32_16x16x4_f32(
                false, a, false, b, (short)0, acc, false, false);
        }
    }

    // Reduce |q|^2 partials: intra-wave butterfly, then LDS atomic per wave
#pragma unroll
    for (int i = 0; i < QT; ++i) {
        float v = sqacc[i];
        for (int o = 16; o > 0; o >>= 1) v += __shfl_xor(v, o, 32);
        if (lane == 0) atomicAdd(&sqq[i], v);
    }

    // Scatter C: VGPR r, lane l -> M = r + (l>=16 ? 8:0), N = l%16
    const int nlo = lane & 15;
    const int mhi = (lane >> 4) << 3;
#pragma unroll
    for (int r = 0; r < 8; ++r)
        distT[(r + mhi) * N_WAY + wave * 16 + nlo] = acc[r];
    __syncthreads();

    // Fused log-softmax over 64 classes; wave w handles rows 4w..4w+3
    const float sqp0 = sqp[lane];
    const float sqp1 = sqp[32 + lane];
    float ceAcc = 0.0f, okAcc = 0.0f;
    for (int rr = 0; rr < 4; ++rr) {
        const int m  = wave * 4 + rr;
        const int gq = qt * QT + m;
        const int y  = gq >> 9;                 // true class (n_query = 512)
        const float sq = sqq[m];
        // x = -dist = (2*dot - |q|^2 - |p|^2)/D
        float x0 = (2.0f * distT[m * N_WAY + lane]      - sq - sqp0) * INV_D;
        float x1 = (2.0f * distT[m * N_WAY + 32 + lane] - sq - sqp1) * INV_D;

        float mx = fmaxf(x0, x1);
        for (int o = 16; o > 0; o >>= 1) mx = fmaxf(mx, __shfl_xor(mx, o, 32));
        float s = __expf(x0 - mx) + __expf(x1 - mx);
        for (int o = 16; o > 0; o >>= 1) s += __shfl_xor(s, o, 32);
        const float ls = __logf(s);
        const float lp0 = x0 - mx - ls;
        const float lp1 = x1 - mx - ls;

        out[(size_t)gq * N_WAY + lane]      = lp0;
        out[(size_t)gq * N_WAY + 32 + lane] = lp1;

        // CE: pick log-prob of true class via masked butterfly-sum
        float ce = (lane == y ? lp0 : 0.0f) + (lane + 32 == y ? lp1 : 0.0f);
        for (int o = 16; o > 0; o >>= 1) ce += __shfl_xor(ce, o, 32);

        // argmax with lowest-index tie-break (matches jnp.argmax)
        float bv = x0; int bi = lane;
        if (x1 > bv) { bv = x1; bi = lane + 32; }
        for (int o = 16; o > 0; o >>= 1) {
            const float ov = __shfl_xor(bv, o, 32);
            const int   oi = __shfl_xor(bi, o, 32);
            if (ov > bv || (ov == bv && oi < bi)) { bv = ov; bi = oi; }
        }
        if (lane == 0) {
            ceAcc += -ce;
            okAcc += (bi == y) ? 1.0f : 0.0f;
        }
    }
    if (lane == 0) { atomicAdd(&blockCE, ceAcc); atomicAdd(&blockAC, okAcc); }
    __syncthreads();
    if (tid == 0) {
        atomicAdd(out + OUT_SCALARS,     blockCE * INV_N);
        atomicAdd(out + OUT_SCALARS + 1, blockAC * INV_N);
    }
}

// ---------------------------------------------------------------------------
extern "C" void kernel_launch(void* const* d_in, const int* in_sizes, int n_in,
                              void* d_out, int out_size, void* d_ws, size_t ws_size,
                              hipStream_t stream) {
    const float* s_emb = (const float*)d_in[0];   // [64,16,2048]
    const float* q_emb = (const float*)d_in[1];   // [64,512,2048]
    float* out = (float*)d_out;                   // 2097152 log-probs + ce + acc

    float* protoB = (float*)d_ws;                 // 64*2048 floats (512 KB)
    float* sqp    = protoB + N_WAY * DIM;         // 64 floats

    hipLaunchKernelGGL(init_kernel, dim3(1), dim3(64), 0, stream, out, sqp);
    hipLaunchKernelGGL(proto_kernel, dim3(512), dim3(256), 0, stream,
                       s_emb, protoB, sqp);
    hipLaunchKernelGGL(proto_dist_kernel, dim3(NQ_TOTAL / QT), dim3(128), 0,
                       stream, q_emb, protoB, sqp, out);
}